// ConvNN_91036126806120
// MI455X (gfx1250) — compile-verified
//
#include <hip/hip_runtime.h>

typedef float v2f __attribute__((ext_vector_type(2)));
typedef float v8f __attribute__((ext_vector_type(8)));

#define B_DIM   1024
#define GROUPS  2048
#define D       64          // IN_DIM == D1 == D2 == 64
#define LSTR    68          // LDS row stride (64 + 4 pad floats) -> conflict-free
#define NEG_SLOPE 0.2f

__global__ __launch_bounds__(256)
void convnn_fused_wmma(const float* __restrict__ x,
                       const float* __restrict__ W0,
                       const float* __restrict__ b0,
                       const float* __restrict__ W1,
                       const float* __restrict__ b1,
                       float* __restrict__ out)
{
    __shared__ float sW0[D * LSTR];
    __shared__ float sW1[D * LSTR];
    __shared__ float sH [D * LSTR];

    const int bt   = blockIdx.x;        // batch tile (64 rows each), 0..15
    const int g    = blockIdx.y;        // group, 0..2047
    const int tid  = threadIdx.x;       // 0..255
    const int lane = tid & 31;
    const int wave = tid >> 5;          // 0..7

    // ---- stage W0_g, W1_g into LDS (coalesced b128 loads) ----
    const float* gW0 = W0 + (size_t)g * (D * D);
    const float* gW1 = W1 + (size_t)g * (D * D);
    #pragma unroll
    for (int p = 0; p < 4; ++p) {
        const int e = (p * 256 + tid) * 4;       // element index 0..4092 (4 stays in-row)
        const int r = e >> 6, c = e & 63;
        const float4 w0 = *(const float4*)(gW0 + e);
        const float4 w1 = *(const float4*)(gW1 + e);
        *(float4*)(&sW0[r * LSTR + c]) = w0;
        *(float4*)(&sW1[r * LSTR + c]) = w1;
    }
    __syncthreads();

    // ---- wave tile assignment: rows tile mt (16 rows), col tiles nt0, nt0+1 ----
    const int mt    = wave & 3;
    const int nt0   = (wave >> 2) << 1;          // 0 or 2
    const int l15   = lane & 15;
    const int khalf = (lane >> 4) << 1;          // K sub-offset: 0 (lanes 0-15) / 2 (lanes 16-31)
    const int hioff = (lane >> 4) << 3;          // C/D row offset: 0 / 8

    const int m0g = bt * 64 + mt * 16;           // global batch row base of this wave's tile
    const int n0a = nt0 * 16;
    const int n0b = n0a + 16;

    // ================= Phase 1: H = leakyrelu(X * W0_g^T + b0_g) =================
    v8f c0, c1;
    {
        const float bva = b0[g * D + n0a + l15]; // bias depends only on N == lane&15
        const float bvb = b0[g * D + n0b + l15];
        #pragma unroll
        for (int r = 0; r < 8; ++r) { c0[r] = bva; c1[r] = bvb; }
    }
    const float* xrow = x + (size_t)(m0g + l15) * D + khalf;      // A frags (L2-resident)
    const float* w0ra = &sW0[(n0a + l15) * LSTR + khalf];         // B frags from LDS
    const float* w0rb = &sW0[(n0b + l15) * LSTR + khalf];
    #pragma unroll
    for (int kb = 0; kb < 16; ++kb) {            // K = 64 = 16 x (k-depth 4)
        const v2f a  = *(const v2f*)(xrow + kb * 4);
        const v2f ba = *(const v2f*)(w0ra + kb * 4);
        const v2f bb = *(const v2f*)(w0rb + kb * 4);
        c0 = __builtin_amdgcn_wmma_f32_16x16x4_f32(false, a, false, ba, (short)0, c0, false, false);
        c1 = __builtin_amdgcn_wmma_f32_16x16x4_f32(false, a, false, bb, (short)0, c1, false, false);
    }
    // LeakyReLU + write H tiles to LDS
    {
        const int mrow = mt * 16 + hioff;
        #pragma unroll
        for (int r = 0; r < 8; ++r) {
            float h0 = c0[r]; h0 = (h0 > 0.0f) ? h0 : NEG_SLOPE * h0;
            float h1 = c1[r]; h1 = (h1 > 0.0f) ? h1 : NEG_SLOPE * h1;
            sH[(mrow + r) * LSTR + n0a + l15] = h0;
            sH[(mrow + r) * LSTR + n0b + l15] = h1;
        }
    }
    __syncthreads();

    // ================= Phase 2: OUT = H * W1_g^T + b1_g =================
    {
        const float bva = b1[g * D + n0a + l15];
        const float bvb = b1[g * D + n0b + l15];
        #pragma unroll
        for (int r = 0; r < 8; ++r) { c0[r] = bva; c1[r] = bvb; }
    }
    const float* hrow = &sH[(mt * 16 + l15) * LSTR + khalf];
    const float* w1ra = &sW1[(n0a + l15) * LSTR + khalf];
    const float* w1rb = &sW1[(n0b + l15) * LSTR + khalf];
    #pragma unroll
    for (int kb = 0; kb < 16; ++kb) {
        const v2f a  = *(const v2f*)(hrow + kb * 4);
        const v2f ba = *(const v2f*)(w1ra + kb * 4);
        const v2f bb = *(const v2f*)(w1rb + kb * 4);
        c0 = __builtin_amdgcn_wmma_f32_16x16x4_f32(false, a, false, ba, (short)0, c0, false, false);
        c1 = __builtin_amdgcn_wmma_f32_16x16x4_f32(false, a, false, bb, (short)0, c1, false, false);
    }
    // stream result tiles: out[b][g][j], b = m0g + hioff + r, j = n0{a,b} + l15
    {
        const size_t rowpitch = (size_t)GROUPS * D;               // 131072 floats per batch row
        float* obase = out + (size_t)(m0g + hioff) * rowpitch + (size_t)g * D;
        #pragma unroll
        for (int r = 0; r < 8; ++r) {
            float* orow = obase + (size_t)r * rowpitch;
            orow[n0a + l15] = c0[r];
            orow[n0b + l15] = c1[r];
        }
    }
}

extern "C" void kernel_launch(void* const* d_in, const int* in_sizes, int n_in,
                              void* d_out, int out_size, void* d_ws, size_t ws_size,
                              hipStream_t stream) {
    const float* x  = (const float*)d_in[0];
    const float* W0 = (const float*)d_in[1];
    const float* b0 = (const float*)d_in[2];
    const float* W1 = (const float*)d_in[3];
    const float* b1 = (const float*)d_in[4];
    float* out = (float*)d_out;

    dim3 grid(B_DIM / 64, GROUPS);   // x = batch tile (fastest -> 16 blocks share a group's weights in L2)
    dim3 block(256);                 // 8 wave32
    convnn_fused_wmma<<<grid, block, 0, stream>>>(x, W0, b0, W1, b1, out);
}